// FLC_Pooling_60516089200681
// MI455X (gfx1250) — compile-verified
//
#include <hip/hip_runtime.h>
#include <hip/hip_bf16.h>

typedef float v2f __attribute__((ext_vector_type(2)));
typedef float v8f __attribute__((ext_vector_type(8)));

#define HIN   224
#define HOUT  112
#define NIMG  1024          // 16 * 64
#define MBLK  7             // 112 / 16 output row blocks
#define KT    56            // 224 / 4 k-steps per GEMM
#define KPACK_ELEMS (MBLK * KT * 64)   // [tile7][kt56][lane32][2] floats

// Packed WMMA operand images of the complex DFT-crop-IDFT kernel K (112x224).
// Slot (t, kt, lane, c) holds K[t*16 + (lane&15)][kt*4 + (lane>>4)*2 + c],
// which is simultaneously the A-operand layout for stage 1 (K as LHS) and the
// B-operand layout for stage 2 (K^T as RHS) of v_wmma_f32_16x16x4_f32.
__device__ __align__(16) float gKr[KPACK_ELEMS];
__device__ __align__(16) float gKi[KPACK_ELEMS];

// Closed form of K(m,p) = (1/224) * sum_{f=-56}^{55} exp(i*pi*f*(2m-p)/112).
// d = 2m-p even: 0.5 at d==0, else 0.  d odd: Dirichlet-type kernel.
__device__ __forceinline__ void Kmp(int m, int p, float& re, float& im) {
  int d = 2 * m - p;
  if (d == 0)       { re = 0.5f; im = 0.0f; return; }
  if ((d & 1) == 0) { re = 0.0f; im = 0.0f; return; }
  const float PI = 3.14159265358979323846f;
  float theta = PI * (float)d / 112.0f;
  float inv   = 1.0f / (224.0f * sinf(0.5f * theta));
  float a     = 56.5f * theta;
  re = sinf(a) * inv;
  im = cosf(a) * inv;
}

__global__ void flc_precompute_kpack() {
  int g = blockIdx.x * blockDim.x + threadIdx.x;
  if (g >= MBLK * KT * 32) return;
  int lane = g & 31;
  int rest = g >> 5;
  int kt   = rest % KT;
  int t    = rest / KT;
  int l15  = lane & 15;
  int lh   = lane >> 4;
  int row  = t * 16 + l15;          // output-row (A) / output-col (B) index
  int kb   = kt * 4 + lh * 2;       // k index base
#pragma unroll
  for (int c = 0; c < 2; ++c) {
    float re, im;
    Kmp(row, kb + c, re, im);
    gKr[g * 2 + c] = re;
    gKi[g * 2 + c] = im;
  }
}

// One workgroup = (image, 16-row output block). 14 waves (448 threads).
// Stage 1: T(16x224, complex) = K[mb rows] * x       (14 column tiles, 1/wave)
// Stage 2: out(16x112)       = Tr*Kr^T + (-Ti)*Ki^T  (7 tiles, waves 0..6)
__global__ void __launch_bounds__(448)
flc_pool_kernel(const float* __restrict__ x, float* __restrict__ out) {
  __shared__ float Tr[16][232];   // padded stride vs 64-bank LDS
  __shared__ float Ti[16][232];   // stores -Imag(T) (pre-negated for stage 2)

  int bx   = blockIdx.x;
  int img  = bx / MBLK;
  int mb   = bx % MBLK;
  int wave = threadIdx.x >> 5;
  int lane = threadIdx.x & 31;
  int l15  = lane & 15;
  int lh   = lane >> 4;

  const float* xim = x + (size_t)img * (HIN * HIN);

  // ---------------- stage 1 ----------------
  {
    int n0 = wave * 16;
    v8f accR = {0.f,0.f,0.f,0.f,0.f,0.f,0.f,0.f};
    v8f accI = {0.f,0.f,0.f,0.f,0.f,0.f,0.f,0.f};
    const float* pKr  = gKr + mb * (KT * 64) + lane * 2;
    const float* pKi  = gKi + mb * (KT * 64) + lane * 2;
    const float* xcol = xim + (lh * 2) * HIN + n0 + l15;
#pragma unroll 4
    for (int kt = 0; kt < KT; ++kt) {
      v2f ar = *(const v2f*)(pKr + kt * 64);        // A = Kr rows (global_load_b64)
      v2f ai = *(const v2f*)(pKi + kt * 64);        // A = Ki rows
      const float* xr = xcol + (kt * 4) * HIN;
      v2f b;                                        // B = x[k0..k0+3][n0..n0+15]
      b.x = xr[0];
      b.y = xr[HIN];
      accR = __builtin_amdgcn_wmma_f32_16x16x4_f32(false, ar, false, b,
                                                   (short)0, accR, false, false);
      accI = __builtin_amdgcn_wmma_f32_16x16x4_f32(false, ai, false, b,
                                                   (short)0, accI, false, false);
    }
#pragma unroll
    for (int j = 0; j < 8; ++j) {
      Tr[lh * 8 + j][n0 + l15] =  accR[j];
      Ti[lh * 8 + j][n0 + l15] = -accI[j];          // fold the minus sign here
    }
  }
  __syncthreads();

  // ---------------- stage 2 ----------------
  if (wave < 7) {                                   // wave-uniform: EXEC stays full
    int n0 = wave * 16;
    v8f acc = {0.f,0.f,0.f,0.f,0.f,0.f,0.f,0.f};
    const float* pKr = gKr + wave * (KT * 64) + lane * 2;   // B = Kr^T tile
    const float* pKi = gKi + wave * (KT * 64) + lane * 2;   // B = Ki^T tile
#pragma unroll 4
    for (int kt = 0; kt < KT; ++kt) {
      int kb = kt * 4 + lh * 2;
      v2f ar = *(const v2f*)&Tr[l15][kb];           // A from LDS (ds_load_b64)
      v2f ai = *(const v2f*)&Ti[l15][kb];
      v2f br = *(const v2f*)(pKr + kt * 64);
      v2f bi = *(const v2f*)(pKi + kt * 64);
      acc = __builtin_amdgcn_wmma_f32_16x16x4_f32(false, ar, false, br,
                                                  (short)0, acc, false, false);
      acc = __builtin_amdgcn_wmma_f32_16x16x4_f32(false, ai, false, bi,
                                                  (short)0, acc, false, false);
    }
    float* o = out + (size_t)img * (HOUT * HOUT) + (mb * 16) * HOUT + n0;
#pragma unroll
    for (int j = 0; j < 8; ++j)
      o[(lh * 8 + j) * HOUT + l15] = acc[j];
  }
}

extern "C" void kernel_launch(void* const* d_in, const int* in_sizes, int n_in,
                              void* d_out, int out_size, void* d_ws, size_t ws_size,
                              hipStream_t stream) {
  (void)in_sizes; (void)n_in; (void)d_ws; (void)ws_size; (void)out_size;
  const float* x = (const float*)d_in[0];
  float* out = (float*)d_out;

  // Rebuild the packed DFT kernel every launch (deterministic, graph-safe).
  flc_precompute_kpack<<<dim3((MBLK * KT * 32 + 255) / 256), dim3(256), 0, stream>>>();
  // 1024 images x 7 row-blocks, 14 waves each.
  flc_pool_kernel<<<dim3(NIMG * MBLK), dim3(448), 0, stream>>>(x, out);
}